// CompGCNConv_5368709120467
// MI455X (gfx1250) — compile-verified
//
#include <hip/hip_runtime.h>

// CompGCN forward for MI455X (gfx1250, wave32).
//
// Pipeline:
//   K1: zero the scatter accumulator (aliases the first N*128 floats of d_out)
//   K2: per-edge circular correlation + global_atomic_add_f32 scatter
//       (self-loop edges use the all-zero relation row -> msg == 0 -> skipped)
//   K3: out = agg @ weight + bias  via V_WMMA_F32_16X16X4_F32, in-place
//   K4: rel_out = rel_pad @ rel_weight  (same WMMA kernel, zero-padded row 400)
//
// GEMM LDS layout: "paired-K" [k/2][row][k&1] so every WMMA A/B fragment is a
// single even-aligned ds_load_b64 with lanes on consecutive word pairs
// (bank-conflict-free, no v_mov repacking). Full W (64 KB) + A tile (33.3 KB)
// resident in LDS: ~96.5 KB, inside the 320 KB/WGP CDNA5 budget.

typedef __attribute__((ext_vector_type(2))) float v2f;
typedef __attribute__((ext_vector_type(8))) float v8f;

#define DDIM 128
#define PA 130  // word stride per k-pair row of the A tile (pad: staging stores
                // rotate banks; fragment loads stay 8B-aligned since PA is even)
#define PW 256  // word stride per k-pair row of the W tile (128 n * 2 slots)

// ---------------------------------------------------------------- K1: zero
__global__ void zero_f4(float4* __restrict__ p, long long n4) {
  long long i = (long long)blockIdx.x * blockDim.x + threadIdx.x;
  long long stride = (long long)gridDim.x * blockDim.x;
  float4 z = make_float4(0.f, 0.f, 0.f, 0.f);
  for (; i < n4; i += stride) p[i] = z;
}

// ------------------------------------------- K2: correlation + scatter-add
// One wave (32 lanes) per edge; lane owns k = 4*lane .. 4*lane+3.
// r is stored twice in LDS so r[(j+k) mod 128] becomes r2[j+k], j+k < 256.
// Per 4-wide j step: 3 x ds_load_b128 feed 16 FMAs -> VALU(dual-issue)-bound.
__global__ void __launch_bounds__(256)
corr_scatter(const float* __restrict__ x, const int* __restrict__ ei,
             const int* __restrict__ et, const float* __restrict__ rel,
             float* __restrict__ agg, int E) {
  __shared__ float sh[8][DDIM];      // h per wave
  __shared__ float sr[8][2 * DDIM];  // duplicated r per wave
  int w = threadIdx.x >> 5;
  int lane = threadIdx.x & 31;
  int e = blockIdx.x * 8 + w;
  bool valid = (e < E);
  int ec = valid ? e : 0;

  int src = ei[ec];           // edge_index[0][e]
  int dst = ei[E + ec];       // edge_index[1][e]
  int t   = et[ec];           // relation id (< 2*num_rels for real edges)

  float4 hv = ((const float4*)(x + (long long)src * DDIM))[lane];
  float4 rv = ((const float4*)(rel + (long long)t * DDIM))[lane];
  ((float4*)sh[w])[lane] = hv;
  ((float4*)sr[w])[lane] = rv;
  ((float4*)(sr[w] + DDIM))[lane] = rv;
  __syncthreads();

  const float* hs = sh[w];
  const float* rs = sr[w] + lane * 4;  // 16B-aligned base for this lane's k
  float a0 = 0.f, a1 = 0.f, a2 = 0.f, a3 = 0.f;
#pragma unroll 8
  for (int j = 0; j < DDIM; j += 4) {
    float4 h4 = *(const float4*)(hs + j);
    float4 ra = *(const float4*)(rs + j);
    float4 rb = *(const float4*)(rs + j + 4);
    a0 = fmaf(h4.x, ra.x, a0); a0 = fmaf(h4.y, ra.y, a0);
    a0 = fmaf(h4.z, ra.z, a0); a0 = fmaf(h4.w, ra.w, a0);
    a1 = fmaf(h4.x, ra.y, a1); a1 = fmaf(h4.y, ra.z, a1);
    a1 = fmaf(h4.z, ra.w, a1); a1 = fmaf(h4.w, rb.x, a1);
    a2 = fmaf(h4.x, ra.z, a2); a2 = fmaf(h4.y, ra.w, a2);
    a2 = fmaf(h4.z, rb.x, a2); a2 = fmaf(h4.w, rb.y, a2);
    a3 = fmaf(h4.x, ra.w, a3); a3 = fmaf(h4.y, rb.x, a3);
    a3 = fmaf(h4.z, rb.y, a3); a3 = fmaf(h4.w, rb.z, a3);
  }
  if (valid) {
    float* o = agg + (long long)dst * DDIM + lane * 4;
    atomicAdd(o + 0, a0);
    atomicAdd(o + 1, a1);
    atomicAdd(o + 2, a2);
    atomicAdd(o + 3, a3);
  }
}

// --------------------------------- K3/K4: [M x 128] @ [128 x 128] via WMMA
// Block = 256 threads = 8 waves, computes a 64x128 output tile.
// Wave w: rows 16*(w&3)..+15, cols 64*(w>>2)..+63 -> 4 v8f accumulators.
// In-place safe: block stages its own 64 A-rows in LDS before overwriting.
__global__ void __launch_bounds__(256)
gemm128_wmma(const float* __restrict__ A, const float* __restrict__ W,
             const float* __restrict__ bias, float* __restrict__ Out,
             int M_src, int M_dst) {
  // sAp[kp][m*2 + (k&1)], kp = k/2 in 0..63, m in 0..63, stride PA=130
  // sWp[kp][n*2 + (k&1)], kp = k/2 in 0..63, n in 0..127, stride PW=256
  __shared__ float sAp[64 * PA];  // 33.3 KB
  __shared__ float sWp[64 * PW];  // 64 KB   (total ~96.5 KB < 320 KB/WGP)

  int tid = threadIdx.x;
  long long m0 = (long long)blockIdx.x * 64;

  // Stage A tile transposed into paired-k layout (rows >= M_src read as 0).
  // Wave-coherent global float4 reads of one row; paired LDS float2 stores.
  for (int i = tid; i < 64 * 32; i += 256) {
    int m = i >> 5;            // row within tile
    int c4 = (i & 31) * 4;     // k base of this float4
    long long gr = m0 + m;
    float4 v = make_float4(0.f, 0.f, 0.f, 0.f);
    if (gr < M_src) v = ((const float4*)(A + gr * DDIM))[i & 31];
    float* p = sAp + (c4 >> 1) * PA + m * 2;
    *(float2*)p        = make_float2(v.x, v.y);  // k = c4, c4+1
    *(float2*)(p + PA) = make_float2(v.z, v.w);  // k = c4+2, c4+3
  }
  // Stage full W into paired-k layout.
  for (int i = tid; i < DDIM * 32; i += 256) {
    int k = i >> 5;
    int c4 = (i & 31) * 4;
    float4 v = ((const float4*)(W + k * DDIM))[i & 31];
    float* p = sWp + (k >> 1) * PW + c4 * 2 + (k & 1);
    p[0] = v.x; p[2] = v.y; p[4] = v.z; p[6] = v.w;
  }
  __syncthreads();

  int w = tid >> 5, lane = tid & 31;
  int msub = w & 3;            // row sub-tile
  int nbase = (w >> 2) * 64;   // column range base
  int lm = lane & 15;
  int g = lane >> 4;           // half-wave group -> which k of the k-quad

  v8f zero = {0.f, 0.f, 0.f, 0.f, 0.f, 0.f, 0.f, 0.f};
  v8f acc[4];
#pragma unroll
  for (int nt = 0; nt < 4; ++nt) acc[nt] = zero;

  const float* aBase = sAp + (msub * 16 + lm) * 2;  // this lane's m slot
  const float* wBase = sWp + (nbase + lm) * 2;      // this lane's n slot

#pragma unroll
  for (int kk = 0; kk < DDIM; kk += 4) {
    int kp = (kk >> 1) + g;  // lane's k-pair row (k = kk+2g, kk+2g+1)
    v2f a = *(const v2f*)(aBase + kp * PA);
    const float* wp = wBase + kp * PW;
#pragma unroll
    for (int nt = 0; nt < 4; ++nt) {
      v2f b = *(const v2f*)(wp + nt * 32);
      acc[nt] = __builtin_amdgcn_wmma_f32_16x16x4_f32(
          false, a, false, b, (short)0, acc[nt], false, false);
    }
  }

  // Epilogue: D layout -> lane holds D[v + 8*g][lm] per n-tile.
#pragma unroll
  for (int nt = 0; nt < 4; ++nt) {
    int col = nbase + nt * 16 + lm;
    float bv = bias ? bias[col] : 0.0f;
#pragma unroll
    for (int v = 0; v < 8; ++v) {
      long long row = m0 + msub * 16 + v + 8 * g;
      if (row < M_dst) Out[row * DDIM + col] = acc[nt][v] + bv;
    }
  }
}

// -------------------------------------------------------------------- host
extern "C" void kernel_launch(void* const* d_in, const int* in_sizes, int n_in,
                              void* d_out, int out_size, void* d_ws,
                              size_t ws_size, hipStream_t stream) {
  const float* x          = (const float*)d_in[0];
  const int*   ei         = (const int*)d_in[1];
  const int*   et         = (const int*)d_in[2];
  const float* rel        = (const float*)d_in[3];
  const float* weight     = (const float*)d_in[4];
  const float* rel_weight = (const float*)d_in[5];
  const float* bias       = (const float*)d_in[6];

  int N     = in_sizes[0] / DDIM;  // 100000 nodes
  int E     = in_sizes[2];         // 600000 real edges
  int NREL2 = in_sizes[3] / DDIM;  // 400 relation rows (pad row 400 is zero)

  float* out_main = (float*)d_out;
  float* rel_out  = out_main + (long long)N * DDIM;

  // K1: zero the scatter accumulator (aliases the 'out' region of d_out).
  long long n4 = (long long)N * (DDIM / 4);
  zero_f4<<<4096, 256, 0, stream>>>((float4*)out_main, n4);

  // K2: correlation + scatter (self-loop messages are identically zero).
  int cblocks = (E + 7) / 8;  // one wave per edge, 8 waves per block
  corr_scatter<<<cblocks, 256, 0, stream>>>(x, ei, et, rel, out_main, E);

  // K3: out = agg @ weight + bias (in-place over the accumulator).
  gemm128_wmma<<<(N + 63) / 64, 256, 0, stream>>>(out_main, weight, bias,
                                                  out_main, N, N);

  // K4: rel_out = rel_pad @ rel_weight (401 rows; row 400 zero via pad guard).
  gemm128_wmma<<<(NREL2 + 1 + 63) / 64, 256, 0, stream>>>(
      rel, rel_weight, nullptr, rel_out, NREL2, NREL2 + 1);
}